// SimpleRoIAlign_16990890623158
// MI455X (gfx1250) — compile-verified
//
#include <hip/hip_runtime.h>
#include <stdint.h>

// SimpleRoIAlign for MI455X (gfx1250, wave32).
// features: [B=2, C=256, H=200, W=304] f32; rois: [R,5] f32 (b,x1,y1,x2,y2)
// out: [R, C, 14, 14] f32. Store-bandwidth bound (~200MB out @ 23.3 TB/s);
// gathers are L2-resident (features 124.5MB < 192MB L2), so we use
// non-temporal stores to keep the output stream from evicting features.

#define OUT_E 14
#define NPTS  (OUT_E * OUT_E)   // 196

#ifndef ROI_ASYNC_ASM
#define ROI_ASYNC_ASM 1         // gfx1250 async load-to-LDS path (ASYNCcnt)
#endif

__global__ __launch_bounds__(224)
void roialign_gather_kernel(const float* __restrict__ features,
                            const float* __restrict__ rois,
                            float* __restrict__ out,
                            const int C, const int H, const int W,
                            const float scale)
{
    const int r   = blockIdx.x;
    const int tid = threadIdx.x;
    const int HW  = H * W;

    __shared__ float s_roi[8];

    // Stage the 5-float RoI record into LDS via the CDNA5 async path.
    if (tid < 5) {
#if ROI_ASYNC_ASM
        const float* g = rois + (size_t)r * 5 + tid;
        // Generic LDS pointer: low 32 bits are the wave-relative LDS byte
        // address (ISA ch.10 aperture rules), which is what VDST wants.
        uint32_t lds_addr = (uint32_t)(uintptr_t)(&s_roi[tid]);
        asm volatile("global_load_async_to_lds_b32 %0, %1, off"
                     :: "v"(lds_addr), "v"(g)
                     : "memory");
#else
        s_roi[tid] = rois[(size_t)r * 5 + tid];
#endif
    }
#if ROI_ASYNC_ASM
#if defined(__has_builtin) && __has_builtin(__builtin_amdgcn_s_wait_asynccnt)
    __builtin_amdgcn_s_wait_asynccnt(0);
#else
    asm volatile("s_wait_asynccnt 0x0" ::: "memory");
#endif
#endif
    __syncthreads();

    if (tid >= NPTS) return;   // 196 of 224 lanes carry points

    const int   b  = (int)s_roi[0];
    const float x1 = s_roi[1], y1 = s_roi[2], x2 = s_roi[3], y2 = s_roi[4];

    // Per-point bilinear setup, entirely in registers.
    const int   pi  = tid / OUT_E;
    const int   pj  = tid - pi * OUT_E;
    const float inv = 1.0f / (float)OUT_E;
    const float sx  = fmaf(((float)pj + 0.5f) * inv, x2 - x1, x1) * scale - 0.5f;
    const float sy  = fmaf(((float)pi + 0.5f) * inv, y2 - y1, y1) * scale - 0.5f;

    const float fx0 = floorf(sx);
    const float fy0 = floorf(sy);
    const float wx1 = sx - fx0, wx0 = 1.0f - wx1;
    const float wy1 = sy - fy0, wy0 = 1.0f - wy1;
    const int ix0 = (int)fx0, iy0 = (int)fy0;
    const int ix1 = ix0 + 1,  iy1 = iy0 + 1;

    const bool vx0 = (ix0 >= 0) && (ix0 < W);
    const bool vx1 = (ix1 >= 0) && (ix1 < W);
    const bool vy0 = (iy0 >= 0) && (iy0 < H);
    const bool vy1 = (iy1 >= 0) && (iy1 < H);

    const int cx0 = min(max(ix0, 0), W - 1);
    const int cx1 = min(max(ix1, 0), W - 1);
    const int cy0 = min(max(iy0, 0), H - 1);
    const int cy1 = min(max(iy1, 0), H - 1);

    const int off00 = cy0 * W + cx0;
    const int off01 = cy0 * W + cx1;
    const int off10 = cy1 * W + cx0;
    const int off11 = cy1 * W + cx1;
    const float w00 = (vy0 && vx0) ? wy0 * wx0 : 0.0f;
    const float w01 = (vy0 && vx1) ? wy0 * wx1 : 0.0f;
    const float w10 = (vy1 && vx0) ? wy1 * wx0 : 0.0f;
    const float w11 = (vy1 && vx1) ? wy1 * wx1 : 0.0f;

    const float* __restrict__ fp = features + (size_t)b * C * HW;
    float*       __restrict__ op = out + (size_t)r * C * NPTS + tid;

    // 256 channel planes: gathers hit L2 (features resident), output streamed
    // with non-temporal hint. Prefetch next plane's rows L2 -> L0.
#pragma unroll 4
    for (int c = 0; c < C; ++c) {
        __builtin_prefetch(fp + HW + off00, 0, 3);   // global_prefetch_b8
        __builtin_prefetch(fp + HW + off10, 0, 3);   // second bilinear row
        const float v = fp[off00] * w00 + fp[off01] * w01
                      + fp[off10] * w10 + fp[off11] * w11;
        __builtin_nontemporal_store(v, op);          // TH=NT store
        fp += HW;
        op += NPTS;
    }
}

extern "C" void kernel_launch(void* const* d_in, const int* in_sizes, int n_in,
                              void* d_out, int out_size, void* d_ws, size_t ws_size,
                              hipStream_t stream) {
    (void)n_in; (void)out_size; (void)d_ws; (void)ws_size;
    const float* features = (const float*)d_in[0];
    const float* rois     = (const float*)d_in[1];
    float*       out      = (float*)d_out;

    const int C = 256, H = 200, W = 304;     // reference shapes
    const int R = in_sizes[1] / 5;           // 1000 RoIs

    roialign_gather_kernel<<<dim3(R), dim3(224), 0, stream>>>(
        features, rois, out, C, H, W, 0.25f);
}